// f_equiformer_net_19782619365626
// MI455X (gfx1250) — compile-verified
//
#include <hip/hip_runtime.h>
#include <hip/hip_bf16.h>
#include <math.h>

// ---------------------------------------------------------------------------
// Problem constants (from reference)
// ---------------------------------------------------------------------------
#define BB   4
#define NCC  4095
#define NN   4096          // NC + 1 (virtual node at index 0)
#define DIN  64
#define DD   128
#define HH   4
#define LL   2
#define KK   6
#define DHH  32            // DD / HH == one wave32 slice
#define NRBF 16
#define HR   16
#define XF   70            // X feature width: 6 + DIM_IN

typedef __attribute__((ext_vector_type(2))) float v2f;
typedef __attribute__((ext_vector_type(8))) float v8f;

// ---------------------------------------------------------------------------
// Workspace layout (float elements)
// ---------------------------------------------------------------------------
static const size_t OFF_IDX  = 0;                         // B*N*K ints
static const size_t OFF_U    = OFF_IDX  + (size_t)BB*NN*KK;        // *3
static const size_t OFF_RBF  = OFF_U    + (size_t)BB*NN*KK*3;      // *16
static const size_t OFF_X0   = OFF_RBF  + (size_t)BB*NN*KK*NRBF;   // B*N*D
static const size_t OFF_X1   = OFF_X0   + (size_t)BB*NN*DD;        // B*N*3*D
static const size_t OFF_H    = OFF_X1   + (size_t)BB*NN*3*DD;      // B*N*D (also feats)
static const size_t OFF_Q    = OFF_H    + (size_t)BB*NN*DD;        // B*N*D (t spans Q+K)
static const size_t OFF_K    = OFF_Q    + (size_t)BB*NN*DD;        // B*N*D
static const size_t OFF_V    = OFF_K    + (size_t)BB*NN*DD;        // B*N*D (also gate)
static const size_t OFF_ATTN = OFF_V    + (size_t)BB*NN*DD;        // B*N*H*K
static const size_t OFF_R1   = OFF_ATTN + (size_t)BB*NN*HH*KK;     // out0 / X1W / stateFull
static const size_t OFF_R2   = OFF_R1   + (size_t)BB*NN*3*DD;      // out1

// ---------------------------------------------------------------------------
// GEMM: C[M x Nd] = act(A[M x Kd] @ B[Kd x Nd] + bias) + res
// one wave -> one 16x16 tile, fp32 WMMA, K stepped by 4.
// act: 0=none, 1=silu, 2=sigmoid
// ---------------------------------------------------------------------------
__global__ __launch_bounds__(256) void gemm_wmma_f32(
    const float* __restrict__ A, const float* __restrict__ Bm,
    const float* bias, const float* res, float* C,
    int M, int Kd, int Nd, int act)
{
  int wv     = blockIdx.x * 8 + (threadIdx.x >> 5);
  int tilesN = Nd >> 4;
  int total  = (M >> 4) * tilesN;
  if (wv >= total) return;
  int tm = wv / tilesN, tn = wv - tm * tilesN;
  int lane = threadIdx.x & 31;
  int half = lane >> 4, l16 = lane & 15;

  v8f acc = {0.f, 0.f, 0.f, 0.f, 0.f, 0.f, 0.f, 0.f};

  // A-fragment: lane (half,l16) holds A[M=l16][k + 2*half + {0,1}]
  // B-fragment: lane (half,l16) holds B[k + 2*half + {0,1}][N=l16]
  const float* Ar = A  + (size_t)(tm * 16 + l16) * Kd + 2 * half;
  const float* Bc = Bm + (size_t)(2 * half) * Nd + tn * 16 + l16;

  for (int k = 0; k < Kd; k += 4) {
    v2f av, bv;
    av.x = Ar[k];
    av.y = Ar[k + 1];
    bv.x = Bc[(size_t)k * Nd];
    bv.y = Bc[(size_t)(k + 1) * Nd];
    acc = __builtin_amdgcn_wmma_f32_16x16x4_f32(false, av, false, bv,
                                                (short)0, acc, false, false);
  }

  int n = tn * 16 + l16;
  float bval = bias ? bias[n] : 0.0f;
#pragma unroll
  for (int v = 0; v < 8; v++) {
    int m = tm * 16 + v + 8 * half;   // C/D layout: M = vgpr + 8*half
    float xv = acc[v] + bval;
    if (act == 1)      xv = xv / (1.0f + __expf(-xv));        // silu
    else if (act == 2) xv = 1.0f / (1.0f + __expf(-xv));      // sigmoid
    if (res) xv += res[(size_t)m * Nd + n];
    C[(size_t)m * Nd + n] = xv;
  }
}

// ---------------------------------------------------------------------------
// KNN (top-6 by squared distance) + unit vectors + RBF features.
// One wave per query node; LDS-tiled candidate positions; per-lane sorted
// top-6 then a 6-round wave-min merge (pop-front, static indexing).
// ---------------------------------------------------------------------------
__global__ __launch_bounds__(256) void knn_build(
    const float* __restrict__ X, int* __restrict__ idxo,
    float* __restrict__ u, float* __restrict__ rbf)
{
  __shared__ float sp[512 * 3];
  int wv = threadIdx.x >> 5, lane = threadIdx.x & 31;
  int gq = blockIdx.x * 8 + wv;         // == b*NN + i
  int b  = gq >> 12;
  int i  = gq & (NN - 1);

  float qx = 0.f, qy = 0.f, qz = 0.f;
  if (i > 0) {
    const float* p = X + ((size_t)b * NCC + (i - 1)) * XF;
    qx = p[0]; qy = p[1]; qz = p[2];
  }

  float ld[KK]; int li[KK];
#pragma unroll
  for (int k = 0; k < KK; k++) { ld[k] = 3.0e38f; li[k] = 0; }

  for (int t = 0; t < NN / 512; t++) {
    for (int s = threadIdx.x; s < 512; s += 256) {
      int j = t * 512 + s;
      float px = 0.f, py = 0.f, pz = 0.f;
      if (j > 0) {
        const float* p = X + ((size_t)b * NCC + (j - 1)) * XF;
        px = p[0]; py = p[1]; pz = p[2];
      }
      sp[s * 3 + 0] = px; sp[s * 3 + 1] = py; sp[s * 3 + 2] = pz;
    }
    __syncthreads();
    for (int s = lane; s < 512; s += 32) {
      int j = t * 512 + s;
      if (j != i) {
        float dx = sp[s * 3] - qx, dy = sp[s * 3 + 1] - qy, dz = sp[s * 3 + 2] - qz;
        float d = dx * dx + dy * dy + dz * dz;
        int jj = j;
#pragma unroll
        for (int k = 0; k < KK; k++) {
          if (d < ld[k]) {
            float td = ld[k]; int tj = li[k];
            ld[k] = d; li[k] = jj; d = td; jj = tj;
          }
        }
      }
    }
    __syncthreads();
  }

  // merge 32 sorted lists -> global top-6
  int sel[KK];
#pragma unroll
  for (int k = 0; k < KK; k++) {
    float myd = ld[0];
    float m = myd;
#pragma unroll
    for (int off = 16; off > 0; off >>= 1) m = fminf(m, __shfl_xor(m, off, 32));
    unsigned long long bal = __ballot(myd == m);
    int src  = (int)__ffsll(bal) - 1;
    int jwin = __shfl(li[0], src, 32);
    sel[k] = jwin;
    if (lane == src) {
#pragma unroll
      for (int s = 0; s < KK - 1; s++) { ld[s] = ld[s + 1]; li[s] = li[s + 1]; }
      ld[KK - 1] = 3.0e38f;
    }
  }

  size_t e0 = (size_t)gq * KK;
  if (lane == 0) {
#pragma unroll
    for (int k = 0; k < KK; k++) idxo[e0 + k] = sel[k];
  }
  if (lane < KK) {
    int j = sel[lane];
    float px = 0.f, py = 0.f, pz = 0.f;
    if (j > 0) {
      const float* p = X + ((size_t)b * NCC + (j - 1)) * XF;
      px = p[0]; py = p[1]; pz = p[2];
    }
    float rx = px - qx, ry = py - qy, rz = pz - qz;
    float dist = sqrtf(rx * rx + ry * ry + rz * rz + 1e-8f);
    float invd = 1.0f / dist;
    size_t e = e0 + lane;
    u[e * 3 + 0] = rx * invd; u[e * 3 + 1] = ry * invd; u[e * 3 + 2] = rz * invd;
#pragma unroll
    for (int r = 0; r < NRBF; r++) {
      float cc = (4.0f / 15.0f) * (float)r;         // linspace(0,4,16)
      float z = (dist - cc) * 4.0f;                 // /0.25
      rbf[e * NRBF + r] = __expf(-z * z);
    }
  }
}

// ---------------------------------------------------------------------------
// feats = [zero row ; X[:,:,6:]]   (input to embed GEMM)
// ---------------------------------------------------------------------------
__global__ void build_feats(const float* __restrict__ X, float* __restrict__ f)
{
  size_t tid = (size_t)blockIdx.x * 256 + threadIdx.x;
  if (tid >= (size_t)BB * NN * DIN) return;
  int d = tid & (DIN - 1);
  size_t r = tid >> 6;
  int n = (int)(r & (NN - 1));
  int b = (int)(r >> 12);
  f[tid] = (n == 0) ? 0.f : X[((size_t)b * NCC + n - 1) * XF + 6 + d];
}

// ---------------------------------------------------------------------------
// LayerNorm: one wave per 128-wide row
// ---------------------------------------------------------------------------
__global__ __launch_bounds__(256) void ln_kernel(
    const float* __restrict__ x, const float* __restrict__ s,
    const float* __restrict__ bbias, float* __restrict__ o, int rows)
{
  int w = blockIdx.x * 8 + (threadIdx.x >> 5);
  if (w >= rows) return;
  int lane = threadIdx.x & 31;
  const float* xr = x + (size_t)w * DD;
  float v[4];
#pragma unroll
  for (int t = 0; t < 4; t++) v[t] = xr[t * 32 + lane];
  float sm = v[0] + v[1] + v[2] + v[3];
  float sq = v[0] * v[0] + v[1] * v[1] + v[2] * v[2] + v[3] * v[3];
#pragma unroll
  for (int off = 16; off > 0; off >>= 1) {
    sm += __shfl_xor(sm, off, 32);
    sq += __shfl_xor(sq, off, 32);
  }
  float mean = sm * (1.0f / DD);
  float var  = sq * (1.0f / DD) - mean * mean;
  float inv  = rsqrtf(var + 1e-5f);
  float* orow = o + (size_t)w * DD;
#pragma unroll
  for (int t = 0; t < 4; t++) {
    int e = t * 32 + lane;
    orow[e] = (v[t] - mean) * inv * s[e] + bbias[e];
  }
}

// ---------------------------------------------------------------------------
// Attention: one wave per (b,n,h); lane == dh (DH==32).
// Recomputes the radial MLP per edge: lanes 0..15 each compute one hidden
// unit, broadcast via shuffles for rv/rg; bias = wave-mean of rg slice.
// ---------------------------------------------------------------------------
__global__ __launch_bounds__(256) void attn_kernel(
    const float* __restrict__ q, const float* __restrict__ kp,
    const float* __restrict__ vv, const int* __restrict__ idx,
    const float* __restrict__ rbf,
    const float* __restrict__ rw1, const float* __restrict__ rb1,
    const float* __restrict__ rw2, const float* __restrict__ rb2,
    float* __restrict__ out0, float* __restrict__ attnb)
{
  int w = blockIdx.x * 8 + (threadIdx.x >> 5);
  if (w >= BB * NN * HH) return;
  int lane = threadIdx.x & 31;
  int h = w & (HH - 1);
  int bn = w >> 2;
  int bbase = (bn >> 12) << 12;
  size_t e0 = (size_t)bn * KK;
  int e = h * DHH + lane;
  float qd = q[(size_t)bn * DD + e];

  float lg[KK], rvk[KK]; int js[KK];
#pragma unroll
  for (int k = 0; k < KK; k++) {
    int j = idx[e0 + k];
    js[k] = j;
    float p = qd * kp[(size_t)(bbase + j) * DD + e];
#pragma unroll
    for (int off = 16; off > 0; off >>= 1) p += __shfl_xor(p, off, 32);
    float logit = p * 0.1767766952966369f;          // 1/sqrt(32)

    const float* rb = rbf + (e0 + k) * NRBF;
    float hid = 0.0f;
    if (lane < HR) {
      float a = rb1[lane];
#pragma unroll
      for (int r = 0; r < NRBF; r++) a = fmaf(rb[r], rw1[r * HR + lane], a);
      hid = a / (1.0f + __expf(-a));                // silu
    }
    float rg = rb2[e];
    float rv = rb2[DD + e];
#pragma unroll
    for (int r = 0; r < HR; r++) {
      float hr = __shfl(hid, r, 32);
      rg = fmaf(hr, rw2[r * 2 * DD + e], rg);
      rv = fmaf(hr, rw2[r * 2 * DD + DD + e], rv);
    }
    float sgm = rg;
#pragma unroll
    for (int off = 16; off > 0; off >>= 1) sgm += __shfl_xor(sgm, off, 32);
    logit += sgm * (1.0f / 32.0f);                  // mean over DH
    lg[k] = logit; rvk[k] = rv;
  }

  float mx = lg[0];
#pragma unroll
  for (int k = 1; k < KK; k++) mx = fmaxf(mx, lg[k]);
  float den = 0.f, ex[KK];
#pragma unroll
  for (int k = 0; k < KK; k++) { ex[k] = __expf(lg[k] - mx); den += ex[k]; }
  float inv = 1.0f / den;

  float acc = 0.f;
#pragma unroll
  for (int k = 0; k < KK; k++) {
    float a  = ex[k] * inv;
    float sv = rvk[k] / (1.0f + __expf(-rvk[k]));   // silu(rv)
    acc = fmaf(a * sv, vv[(size_t)(bbase + js[k]) * DD + e], acc);
    if (lane == 0) attnb[(size_t)w * KK + k] = a;
  }
  out0[(size_t)bn * DD + e] = acc;
}

// ---------------------------------------------------------------------------
// Vector channel combine: out1[b,n,c,:] = sum_k attn[h(e)] *
//    ( X1W[b, idx_k, c, e] + rv[k,e] * u[k,c] );  one wave per (b,n,c).
// ---------------------------------------------------------------------------
__global__ __launch_bounds__(256) void vec_combine(
    const float* __restrict__ x1w, const float* __restrict__ u,
    const float* __restrict__ rbf, const int* __restrict__ idx,
    const float* __restrict__ attnb,
    const float* __restrict__ rw1, const float* __restrict__ rb1,
    const float* __restrict__ rw2, const float* __restrict__ rb2,
    float* __restrict__ out1)
{
  int w = blockIdx.x * 8 + (threadIdx.x >> 5);
  if (w >= BB * NN * 3) return;
  int lane = threadIdx.x & 31;
  int c  = w % 3;
  int bn = w / 3;
  int bbase = (bn >> 12) << 12;
  size_t e0 = (size_t)bn * KK;

  float acc[4] = {0.f, 0.f, 0.f, 0.f};
#pragma unroll
  for (int k = 0; k < KK; k++) {
    int j = idx[e0 + k];
    float uu = u[(e0 + k) * 3 + c];
    const float* rb = rbf + (e0 + k) * NRBF;
    float hid = 0.f;
    if (lane < HR) {
      float a = rb1[lane];
#pragma unroll
      for (int r = 0; r < NRBF; r++) a = fmaf(rb[r], rw1[r * HR + lane], a);
      hid = a / (1.0f + __expf(-a));
    }
    float rv[4];
#pragma unroll
    for (int t = 0; t < 4; t++) rv[t] = rb2[DD + t * 32 + lane];
#pragma unroll
    for (int r = 0; r < HR; r++) {
      float hr = __shfl(hid, r, 32);
#pragma unroll
      for (int t = 0; t < 4; t++)
        rv[t] = fmaf(hr, rw2[r * 2 * DD + DD + t * 32 + lane], rv[t]);
    }
    const float* xw = x1w + ((size_t)(bbase + j) * 3 + c) * DD;
#pragma unroll
    for (int t = 0; t < 4; t++) {                 // t == head index (e = 32t+lane)
      float a = attnb[((size_t)bn * HH + t) * KK + k];
      acc[t] = fmaf(a, xw[t * 32 + lane] + rv[t] * uu, acc[t]);
    }
  }
  float* o = out1 + ((size_t)bn * 3 + c) * DD;
#pragma unroll
  for (int t = 0; t < 4; t++) o[t * 32 + lane] = acc[t];
}

// ---------------------------------------------------------------------------
// x1 *= sigmoid-gate (broadcast over the 3 spatial components)
// ---------------------------------------------------------------------------
__global__ void gate_mul(float* __restrict__ x1, const float* __restrict__ g)
{
  size_t tid = (size_t)blockIdx.x * 256 + threadIdx.x;
  if (tid >= (size_t)BB * NN * 3 * DD) return;
  int e = (int)(tid & (DD - 1));
  size_t bn = (tid >> 7) / 3;
  x1[tid] *= g[bn * DD + e];
}

// ---------------------------------------------------------------------------
// Heads
// ---------------------------------------------------------------------------
__global__ void pack_state(const float* __restrict__ sf, float* __restrict__ o)
{
  size_t tid = (size_t)blockIdx.x * 256 + threadIdx.x;
  if (tid >= (size_t)BB * NCC * DIN) return;
  int d = (int)(tid & (DIN - 1));
  size_t r = tid >> 6;
  int n = (int)(r % NCC);
  int b = (int)(r / NCC);
  o[tid] = sf[((size_t)b * NN + n + 1) * DIN + d];
}

__global__ __launch_bounds__(256) void force_kernel(
    const float* __restrict__ x1, const float* __restrict__ wf,
    const float* __restrict__ bf, float* __restrict__ fo)
{
  int w = blockIdx.x * 8 + (threadIdx.x >> 5);
  if (w >= BB * NCC * 3) return;
  int lane = threadIdx.x & 31;
  int c = w % 3;
  int r = w / 3;
  int n = r % NCC;
  int b = r / NCC;
  const float* xr = x1 + (((size_t)b * NN + n + 1) * 3 + c) * DD;
  float p = 0.f;
#pragma unroll
  for (int t = 0; t < 4; t++) p = fmaf(xr[t * 32 + lane], wf[t * 32 + lane], p);
#pragma unroll
  for (int off = 16; off > 0; off >>= 1) p += __shfl_xor(p, off, 32);
  if (lane == 0) fo[((size_t)b * NCC + n) * 3 + c] = p + bf[0];
}

__global__ void force_center(float* __restrict__ f)
{
  int b = blockIdx.x / 3, c = blockIdx.x % 3;
  __shared__ float red[256];
  float s = 0.f;
  for (int n = threadIdx.x; n < NCC; n += 256) s += f[((size_t)b * NCC + n) * 3 + c];
  red[threadIdx.x] = s;
  __syncthreads();
  for (int st = 128; st > 0; st >>= 1) {
    if (threadIdx.x < st) red[threadIdx.x] += red[threadIdx.x + st];
    __syncthreads();
  }
  float m = red[0] * (1.0f / NCC);
  for (int n = threadIdx.x; n < NCC; n += 256) f[((size_t)b * NCC + n) * 3 + c] -= m;
}

// ---------------------------------------------------------------------------
// Host driver
// ---------------------------------------------------------------------------
extern "C" void kernel_launch(void* const* d_in, const int* in_sizes, int n_in,
                              void* d_out, int out_size, void* d_ws, size_t ws_size,
                              hipStream_t stream)
{
  (void)in_sizes; (void)n_in; (void)out_size; (void)ws_size;
  const float* X       = (const float*)d_in[0];
  const float* W_embed = (const float*)d_in[1];
  const float* b_embed = (const float*)d_in[2];
  const float* Wq  = (const float*)d_in[3];
  const float* Wk  = (const float*)d_in[4];
  const float* Wv  = (const float*)d_in[5];
  const float* Wvv = (const float*)d_in[6];
  const float* rw1 = (const float*)d_in[7];
  const float* rb1 = (const float*)d_in[8];
  const float* rw2 = (const float*)d_in[9];
  const float* rb2 = (const float*)d_in[10];
  const float* Wo  = (const float*)d_in[11];
  const float* Wov = (const float*)d_in[12];
  const float* ffw1= (const float*)d_in[13];
  const float* ffb1= (const float*)d_in[14];
  const float* ffw2= (const float*)d_in[15];
  const float* ffb2= (const float*)d_in[16];
  const float* gw  = (const float*)d_in[17];
  const float* gb  = (const float*)d_in[18];
  const float* l1s = (const float*)d_in[19];
  const float* l1b = (const float*)d_in[20];
  const float* l2s = (const float*)d_in[21];
  const float* l2b = (const float*)d_in[22];
  const float* Ws  = (const float*)d_in[23];
  const float* bs  = (const float*)d_in[24];
  const float* Wf  = (const float*)d_in[25];
  const float* bfc = (const float*)d_in[26];

  float* ws   = (float*)d_ws;
  int*   idx  = (int*)(ws + OFF_IDX);
  float* u    = ws + OFF_U;
  float* rbf  = ws + OFF_RBF;
  float* x0   = ws + OFF_X0;
  float* x1   = ws + OFF_X1;
  float* h    = ws + OFF_H;
  float* qb   = ws + OFF_Q;
  float* kb   = ws + OFF_K;
  float* vb   = ws + OFF_V;
  float* attn = ws + OFF_ATTN;
  float* r1   = ws + OFF_R1;   // out0 / X1W / stateFull
  float* r2   = ws + OFF_R2;   // out1
  float* tbuf = qb;            // FFN hidden (B*N*256) spans Q+K regions
  float* gbuf = vb;            // gate reuses V region

  float* out_state = (float*)d_out;
  float* out_force = out_state + (size_t)BB * NCC * DIN;

  auto gemm = [&](const float* A, const float* Bm, const float* bias,
                  const float* res, float* C, int M, int Kd, int Nd, int act) {
    int waves  = (M >> 4) * (Nd >> 4);
    int blocks = (waves + 7) / 8;
    gemm_wmma_f32<<<blocks, 256, 0, stream>>>(A, Bm, bias, res, C, M, Kd, Nd, act);
  };

  const int ROWS  = BB * NN;        // 16384
  const int ROWS3 = BB * NN * 3;    // 49152

  // x1 = 0
  hipMemsetAsync(x1, 0, (size_t)ROWS3 * DD * sizeof(float), stream);

  // KNN + edge geometry + RBF
  knn_build<<<ROWS / 8, 256, 0, stream>>>(X, idx, u, rbf);

  // x0 = feats @ W_embed + b_embed
  build_feats<<<(int)(((size_t)ROWS * DIN + 255) / 256), 256, 0, stream>>>(X, h);
  gemm(h, W_embed, b_embed, nullptr, x0, ROWS, DIN, DD, 0);

  for (int l = 0; l < LL; l++) {
    const float* Wq_l  = Wq  + (size_t)l * DD * DD;
    const float* Wk_l  = Wk  + (size_t)l * DD * DD;
    const float* Wv_l  = Wv  + (size_t)l * DD * DD;
    const float* Wvv_l = Wvv + (size_t)l * DD * DD;
    const float* Wo_l  = Wo  + (size_t)l * DD * DD;
    const float* Wov_l = Wov + (size_t)l * DD * DD;
    const float* rw1_l = rw1 + (size_t)l * NRBF * HR;
    const float* rb1_l = rb1 + (size_t)l * HR;
    const float* rw2_l = rw2 + (size_t)l * HR * 2 * DD;
    const float* rb2_l = rb2 + (size_t)l * 2 * DD;
    const float* fw1_l = ffw1 + (size_t)l * DD * 2 * DD;
    const float* fb1_l = ffb1 + (size_t)l * 2 * DD;
    const float* fw2_l = ffw2 + (size_t)l * 2 * DD * DD;
    const float* fb2_l = ffb2 + (size_t)l * DD;
    const float* gw_l  = gw  + (size_t)l * DD * DD;
    const float* gb_l  = gb  + (size_t)l * DD;

    // h = LN1(x0)
    ln_kernel<<<ROWS / 8, 256, 0, stream>>>(x0, l1s + l * DD, l1b + l * DD, h, ROWS);

    // q/k/v projections
    gemm(h, Wq_l, nullptr, nullptr, qb, ROWS, DD, DD, 0);
    gemm(h, Wk_l, nullptr, nullptr, kb, ROWS, DD, DD, 0);
    gemm(h, Wv_l, nullptr, nullptr, vb, ROWS, DD, DD, 0);

    // attention -> out0 (in r1), attn weights
    attn_kernel<<<(ROWS * HH) / 8, 256, 0, stream>>>(
        qb, kb, vb, idx, rbf, rw1_l, rb1_l, rw2_l, rb2_l, r1, attn);

    // x0 += out0 @ Wo
    gemm(r1, Wo_l, nullptr, x0, x0, ROWS, DD, DD, 0);

    // X1W = x1 @ Wvv  (gather commutes with the linear map)
    gemm(x1, Wvv_l, nullptr, nullptr, r1, ROWS3, DD, DD, 0);

    // out1 = sum_k attn * (gather(X1W) + rv * u)
    vec_combine<<<ROWS3 / 8, 256, 0, stream>>>(
        r1, u, rbf, idx, attn, rw1_l, rb1_l, rw2_l, rb2_l, r2);

    // x1 += out1 @ Wov
    gemm(r2, Wov_l, nullptr, x1, x1, ROWS3, DD, DD, 0);

    // h = LN2(x0)
    ln_kernel<<<ROWS / 8, 256, 0, stream>>>(x0, l2s + l * DD, l2b + l * DD, h, ROWS);

    // FFN: t = silu(h @ ff_w1 + b1); x0 += t @ ff_w2 + b2
    gemm(h, fw1_l, fb1_l, nullptr, tbuf, ROWS, DD, 2 * DD, 1);
    // gate = sigmoid(h @ gate_w + gate_b)  (before tbuf/x0 updates touch h? h untouched)
    gemm(h, gw_l, gb_l, nullptr, gbuf, ROWS, DD, DD, 2);
    gemm(tbuf, fw2_l, fb2_l, x0, x0, ROWS, 2 * DD, DD, 0);

    // x1 *= gate (broadcast over c)
    gate_mul<<<(int)(((size_t)ROWS3 * DD + 255) / 256), 256, 0, stream>>>(x1, gbuf);
  }

  // state head: full [B,N,64] into r1 then pack rows 1..N-1
  gemm(x0, Ws, bs, nullptr, r1, ROWS, DD, DIN, 0);
  pack_state<<<(int)(((size_t)BB * NCC * DIN + 255) / 256), 256, 0, stream>>>(r1, out_state);

  // forces: per-(b,n,c) dot with W_force, then mean-center over nodes
  force_kernel<<<(BB * NCC * 3 + 7) / 8, 256, 0, stream>>>(x1, Wf, bfc, out_force);
  force_center<<<BB * 3, 256, 0, stream>>>(out_force);
}